// lightGCN_85083302134066
// MI455X (gfx1250) — compile-verified
//
#include <hip/hip_runtime.h>
#include <hip/hip_bf16.h>

typedef __attribute__((ext_vector_type(2))) float v2f;
typedef __attribute__((ext_vector_type(8))) float v8f;

#define D_FEAT 64
#define N_CLASS 2

// ---------------------------------------------------------------------------
// Zero workspace (h1 and h2) with wide stores.
// ---------------------------------------------------------------------------
__global__ void lgcn_zero_kernel(float4* __restrict__ p, int n4) {
    int i = blockIdx.x * blockDim.x + threadIdx.x;
    int stride = gridDim.x * blockDim.x;
    for (; i < n4; i += stride) {
        p[i] = make_float4(0.f, 0.f, 0.f, 0.f);
    }
}

// ---------------------------------------------------------------------------
// COO SpMM: out[dst[e]] += val[e] * x[src[e]], F=64.
// One wave per edge iteration: 32 lanes x float2 = one 64-feature row,
// perfectly coalesced 256B gather + hardware fp32 atomics for the scatter.
// ---------------------------------------------------------------------------
__global__ void lgcn_spmm_kernel(const int* __restrict__ esrc,
                                 const int* __restrict__ edst,
                                 const float* __restrict__ eval,
                                 const float* __restrict__ x,
                                 float* __restrict__ out,
                                 int n_edges) {
    const int lane = threadIdx.x & 31;
    const int wave = (blockIdx.x * blockDim.x + threadIdx.x) >> 5;
    const int nwaves = (gridDim.x * blockDim.x) >> 5;

    for (int e = wave; e < n_edges; e += nwaves) {
        const int s = esrc[e];
        const int d = edst[e];
        const float v = eval[e];

        // Prefetch next edge's source row (global_prefetch_b8).
        const int en = e + nwaves;
        if (en < n_edges) {
            const int sn = esrc[en];
            __builtin_prefetch(&x[(size_t)sn * D_FEAT + lane * 2], 0, 1);
        }

        const v2f xv = *(const v2f*)(&x[(size_t)s * D_FEAT + lane * 2]);
        float* o = &out[(size_t)d * D_FEAT + lane * 2];
        unsafeAtomicAdd(o + 0, v * xv.x);   // global_atomic_add_f32
        unsafeAtomicAdd(o + 1, v * xv.y);
    }
}

// ---------------------------------------------------------------------------
// FC ([N,64] @ [64,2]^T + b) + 2-class log_softmax using fp32 WMMA.
// One wave computes a 16-node tile: D(16x16) = A(16x64-node-feats) x
// B(64x16, cols 0..1 = fc_w rows, rest zero), K in 16 steps of 4 via
// V_WMMA_F32_16X16X4_F32 (keeps everything fp32-exact).
//
// Layouts (wave32):
//   A step kk: lane L holds A[M = L&15][K = kk + (L>>4)*2 + {0,1}] in {a.x,a.y}
//   B step kk: lane L holds B[K = kk + (L>>4)*2 + {0,1}][N = L&15] in {b.x,b.y}
//   D: VGPR r, lanes 0-15 -> D[M=r][N=lane]; lanes 16-31 -> D[M=r+8][N=lane-16]
// ---------------------------------------------------------------------------
__global__ void lgcn_fc_logsoftmax_kernel(const float* __restrict__ h,
                                          const float* __restrict__ w,   // [2][64]
                                          const float* __restrict__ bias,// [2]
                                          float* __restrict__ out,       // [N][2]
                                          int n_nodes) {
    const int lane = threadIdx.x & 31;
    const int wave = (blockIdx.x * blockDim.x + threadIdx.x) >> 5;
    const int m0 = wave * 16;
    if (m0 >= n_nodes) return;           // wave-uniform -> EXEC stays all-1s

    const int row   = lane & 15;         // A: M row; B/D: N column
    const int khalf = (lane >> 4) * 2;   // 0 or 2
    const int wr    = (row < N_CLASS) ? row : 0;  // clamped index, no divergence

    const float* arow = h + (size_t)(m0 + row) * D_FEAT + khalf;

    v8f c = {};
#pragma unroll
    for (int kk = 0; kk < D_FEAT; kk += 4) {
        v2f a = *(const v2f*)(arow + kk);          // 8B-aligned float2
        float bx = w[wr * D_FEAT + kk + khalf + 0];
        float by = w[wr * D_FEAT + kk + khalf + 1];
        v2f b;
        b.x = (row < N_CLASS) ? bx : 0.f;
        b.y = (row < N_CLASS) ? by : 0.f;
        c = __builtin_amdgcn_wmma_f32_16x16x4_f32(
            /*neg_a=*/false, a, /*neg_b=*/false, b,
            /*c_mod=*/(short)0, c, /*reuse_a=*/false, /*reuse_b=*/false);
    }

    const float bv = (row < N_CLASS) ? bias[wr] : 0.f;

#pragma unroll
    for (int r = 0; r < 8; ++r) {
        float l = c[r] + bv;                     // logit for (node m0+M, class=row)
        float other = __shfl_xor(l, 1, 32);      // swap class 0 <-> class 1
        float mx  = fmaxf(l, other);
        float lse = mx + __logf(__expf(l - mx) + __expf(other - mx));
        float res = l - lse;

        const int M = r + ((lane >> 4) << 3);    // rows r (lo half) / r+8 (hi half)
        const int node = m0 + M;
        if (row < N_CLASS && node < n_nodes) {
            out[node * N_CLASS + row] = res;
        }
    }
}

// ---------------------------------------------------------------------------
// Launch: zero(h1,h2) -> spmm(x->h1) -> spmm(h1->h2) -> wmma FC+log_softmax
// ---------------------------------------------------------------------------
extern "C" void kernel_launch(void* const* d_in, const int* in_sizes, int n_in,
                              void* d_out, int out_size, void* d_ws, size_t ws_size,
                              hipStream_t stream) {
    const float* x    = (const float*)d_in[0];
    const int*   esrc = (const int*)d_in[1];
    const int*   edst = (const int*)d_in[2];
    const float* eval = (const float*)d_in[3];
    const float* fcw  = (const float*)d_in[4];
    const float* fcb  = (const float*)d_in[5];
    float* out = (float*)d_out;

    const int n_nodes = in_sizes[0] / D_FEAT;
    const int n_edges = in_sizes[1];

    const size_t h_elems = (size_t)n_nodes * D_FEAT;
    if (ws_size < 2 * h_elems * sizeof(float)) return;

    float* h1 = (float*)d_ws;
    float* h2 = h1 + h_elems;

    // Zero both propagation buffers (2 * N * 64 floats, float4 stores).
    const int n4 = (int)(2 * h_elems / 4);
    lgcn_zero_kernel<<<1024, 256, 0, stream>>>((float4*)d_ws, n4);

    // Two propagation hops (atomic scatter SpMM).
    lgcn_spmm_kernel<<<4096, 256, 0, stream>>>(esrc, edst, eval, x, h1, n_edges);
    lgcn_spmm_kernel<<<4096, 256, 0, stream>>>(esrc, edst, eval, h1, h2, n_edges);

    // FC + log_softmax via fp32 WMMA, 16 nodes per wave.
    const int waves  = (n_nodes + 15) / 16;
    const int blocks = (waves + 7) / 8;          // 256 thr = 8 waves / block
    lgcn_fc_logsoftmax_kernel<<<blocks, 256, 0, stream>>>(h2, fcw, fcb, out, n_nodes);
}